// ContinuousInterpolationFullNetwork_11922829213823
// MI455X (gfx1250) — compile-verified
//
#include <hip/hip_runtime.h>
#include <math.h>

#define NN 384
#define NE 2048
#define NN2 (NN * NN) // 147456

typedef float v2f __attribute__((ext_vector_type(2)));
typedef float v8f __attribute__((ext_vector_type(8)));

// ---------------------------------------------------------------------------
// Prep kernels (deterministic fixed-order scans; no atomics)
// ---------------------------------------------------------------------------
__global__ void k_dinv(const int* __restrict__ ei, const float* __restrict__ w,
                       float* __restrict__ dinv) {
  int v = threadIdx.x;
  if (v >= NN) return;
  float d = 0.f;
  for (int e = 0; e < NE; ++e)
    if (ei[NE + e] == v) d += w[e];
  dinv[v] = (d > 0.f) ? rsqrtf(d) : 0.f;
}

__global__ void k_norm(const int* __restrict__ ei, const float* __restrict__ w,
                       const float* __restrict__ dinv, float* __restrict__ nrm) {
  int e = blockIdx.x * blockDim.x + threadIdx.x;
  if (e >= NE) return;
  int s = ei[e], d = ei[NE + e];
  nrm[e] = dinv[s] * w[e] * dinv[d];
}

// Dense normalized adjacency: S[dst][src] += norm (row-owner scan, in edge order)
__global__ void k_buildS(const int* __restrict__ ei, const float* __restrict__ nrm,
                         float* __restrict__ S) {
  int v = threadIdx.x;
  if (v >= NN) return;
  float* row = S + v * NN;
  for (int u = 0; u < NN; ++u) row[u] = 0.f;
  for (int e = 0; e < NE; ++e)
    if (ei[NE + e] == v) row[ei[e]] += nrm[e];
}

// Zero-pad a (nk, cin, cout) weight stack to (nk, cinP, coutP)
__global__ void k_pad_w(const float* __restrict__ W, float* __restrict__ Wp,
                        int nk, int cin, int cout, int cinP, int coutP) {
  int i = blockIdx.x * blockDim.x + threadIdx.x;
  int tot = nk * cinP * coutP;
  if (i >= tot) return;
  int co = i % coutP;
  int t = i / coutP;
  int ci = t % cinP;
  int k = t / cinP;
  Wp[i] = (ci < cin && co < cout) ? W[(k * cin + ci) * cout + co] : 0.f;
}

__global__ void k_pad_b(const float* __restrict__ b, float* __restrict__ bp,
                        int c, int cp) {
  int i = blockIdx.x * blockDim.x + threadIdx.x;
  if (i >= cp) return;
  bp[i] = (i < c) ? b[i] : 0.f;
}

// ---------------------------------------------------------------------------
// FP32 WMMA GEMM: C = A@B (+C), row-major, lda=K ldb=N ldc=N.
// Wave tile: (16*MT) x (16*NT). Requires M%(16*MT)==0, N%(16*NT)==0, K%4==0.
// V_WMMA_F32_16X16X4_F32 layouts (ISA 7.12.2):
//   A 16x4 : lane = row m (lane&15), a[v] = A[m][k0 + v + 2*(lane>>4)]
//   B 4x16 : lane = col n (lane&15), b[v] = B[k0 + v + 2*(lane>>4)][n]
//   C 16x16: lane = col n, d[r] = C[r + 8*(lane>>4)][n]
// MT=2 register blocking: each B fragment feeds two m-tiles (2x WMMA per
// loaded B dword). acc = MT*NT*8 = 64 VGPRs.
// ---------------------------------------------------------------------------
template <int MT, int NT, bool ACCUM>
__global__ void k_gemm(const float* __restrict__ A, const float* __restrict__ B,
                       float* __restrict__ C, int M, int N, int K) {
  int lane = threadIdx.x & 31;
  int wave = threadIdx.x >> 5;
  int tilesN = N / (16 * NT);
  int stilesM = M / (16 * MT);
  int tile = blockIdx.x * 4 + wave;          // wave-uniform
  if (tile >= stilesM * tilesN) return;      // uniform exit: EXEC stays all-1s
  int tm0 = (tile / tilesN) * MT;
  int tn = tile % tilesN;
  int l15 = lane & 15;
  int hi = lane >> 4;                        // 0 or 1

  int bcol = tn * 16 * NT + l15;

  v8f acc[MT][NT];
#pragma unroll
  for (int m = 0; m < MT; ++m)
#pragma unroll
    for (int t = 0; t < NT; ++t) {
      if (ACCUM) {
#pragma unroll
        for (int r = 0; r < 8; ++r)
          acc[m][t][r] = C[((tm0 + m) * 16 + r + 8 * hi) * N + bcol + t * 16];
      } else {
#pragma unroll
        for (int r = 0; r < 8; ++r) acc[m][t][r] = 0.f;
      }
    }

  const float* Ap[MT];
#pragma unroll
  for (int m = 0; m < MT; ++m)
    Ap[m] = A + (long)((tm0 + m) * 16 + l15) * K;

  for (int k0 = 0; k0 < K; k0 += 4) {
    v2f a[MT];
#pragma unroll
    for (int m = 0; m < MT; ++m) {
      a[m][0] = Ap[m][k0 + 2 * hi];
      a[m][1] = Ap[m][k0 + 2 * hi + 1];
    }
    const float* Bp = B + (long)(k0 + 2 * hi) * N + bcol;
    v2f b[NT];
#pragma unroll
    for (int t = 0; t < NT; ++t) {
      b[t][0] = Bp[t * 16];
      b[t][1] = Bp[N + t * 16];
    }
#pragma unroll
    for (int m = 0; m < MT; ++m)
#pragma unroll
      for (int t = 0; t < NT; ++t)
        acc[m][t] = __builtin_amdgcn_wmma_f32_16x16x4_f32(
            false, a[m], false, b[t], (short)0, acc[m][t], false, false);
  }

#pragma unroll
  for (int m = 0; m < MT; ++m)
#pragma unroll
    for (int t = 0; t < NT; ++t)
#pragma unroll
      for (int r = 0; r < 8; ++r)
        C[((tm0 + m) * 16 + r + 8 * hi) * N + bcol + t * 16] = acc[m][t][r];
}

static inline void launch_gemm(const float* A, const float* B, float* C, int M,
                               int N, int K, bool accum, hipStream_t s) {
  int NT = (N % 64 == 0) ? 4 : ((N % 32 == 0) ? 2 : 1);
  int tiles = (M / 32) * (N / (16 * NT));   // MT = 2 everywhere (M % 32 == 0)
  int blocks = (tiles + 3) / 4;
  if (NT == 4) {
    if (accum) k_gemm<2, 4, true><<<blocks, 128, 0, s>>>(A, B, C, M, N, K);
    else       k_gemm<2, 4, false><<<blocks, 128, 0, s>>>(A, B, C, M, N, K);
  } else if (NT == 2) {
    if (accum) k_gemm<2, 2, true><<<blocks, 128, 0, s>>>(A, B, C, M, N, K);
    else       k_gemm<2, 2, false><<<blocks, 128, 0, s>>>(A, B, C, M, N, K);
  } else {
    if (accum) k_gemm<2, 1, true><<<blocks, 128, 0, s>>>(A, B, C, M, N, K);
    else       k_gemm<2, 1, false><<<blocks, 128, 0, s>>>(A, B, C, M, N, K);
  }
}

// ---------------------------------------------------------------------------
// Elementwise kernels
// ---------------------------------------------------------------------------
__global__ void k_cf_h0(const float* __restrict__ x, float* __restrict__ H) {
  int i = blockIdx.x * blockDim.x + threadIdx.x;  // (384,16)
  if (i >= NN * 16) return;
  H[i] = ((i & 15) == 0) ? x[i >> 4] : 0.f;
}

__global__ void k_bias_relu(const float* __restrict__ ACC, const float* __restrict__ b,
                            float* __restrict__ H, int rows, int C) {
  int i = blockIdx.x * blockDim.x + threadIdx.x;
  if (i >= rows * C) return;
  float v = ACC[i] + b[i % C];
  H[i] = v > 0.f ? v : 0.f;
}

__global__ void k_sigc(const float* __restrict__ ACC, const float* __restrict__ b,
                       float* __restrict__ c, int C) {
  int v = blockIdx.x * blockDim.x + threadIdx.x;
  if (v >= NN) return;
  float z = ACC[v * C] + b[0];
  c[v] = 1.f / (1.f + expf(-z));
}

// P-net input: h[i,j,ch] with ch0 = (i==j), ch1 = c[i], pad 0. C=16.
__global__ void k_p_h0(const float* __restrict__ c, float* __restrict__ H) {
  int idx = blockIdx.x * blockDim.x + threadIdx.x;
  if (idx >= NN2 * 16) return;
  int ch = idx & 15;
  int t = idx >> 4;
  int j = t % NN;
  int i = t / NN;
  float v = 0.f;
  if (ch == 0) v = (i == j) ? 1.f : 0.f;
  else if (ch == 1) v = c[i];
  H[idx] = v;
}

// Final: Phat = relu(ACC[...,0]+b0); P_masked = Phat * (c[j]>=0.5); C = diag(c)
__global__ void k_final(const float* __restrict__ ACC, const float* __restrict__ b,
                        const float* __restrict__ c, float* __restrict__ out, int C) {
  int idx = blockIdx.x * blockDim.x + threadIdx.x;
  if (idx >= NN2) return;
  int j = idx % NN;
  int i = idx / NN;
  float ph = ACC[idx * C] + b[0];
  ph = ph > 0.f ? ph : 0.f;
  out[idx] = ph * (c[j] >= 0.5f ? 1.f : 0.f);
  out[NN2 + idx] = ph;
  out[2 * NN2 + idx] = (i == j) ? c[i] : 0.f;
}

// ---------------------------------------------------------------------------
// Host orchestration
// ---------------------------------------------------------------------------
extern "C" void kernel_launch(void* const* d_in, const int* in_sizes, int n_in,
                              void* d_out, int out_size, void* d_ws, size_t ws_size,
                              hipStream_t stream) {
  (void)in_sizes; (void)n_in; (void)out_size; (void)ws_size;
  const float* x   = (const float*)d_in[0];
  const int*   ei  = (const int*)d_in[1];
  const float* ew  = (const float*)d_in[2];
  float* out = (float*)d_out;

  // network shapes
  static const int cfCin[5]   = {1, 60, 100, 200, 80};
  static const int cfCout[5]  = {60, 100, 200, 80, 1};
  static const int cfCinP[5]  = {16, 64, 112, 208, 80};
  static const int cfCoutP[5] = {64, 112, 208, 80, 16};
  static const int pCin[4]    = {2, 15, 30, 15};
  static const int pCout[4]   = {15, 30, 15, 1};
  static const int pCinP[4]   = {16, 16, 32, 16};
  static const int pCoutP[4]  = {16, 32, 16, 16};
  const int KCF = 20, KP = 50;

  // workspace layout (floats)
  float* ws    = (float*)d_ws;
  float* dinv  = ws;                    // 384
  float* nrm   = dinv + NN;             // 2048
  float* cvec  = nrm + NE;              // 384
  float* S     = cvec + NN;             // 147456
  float* cfA   = S + NN2;               // 384*208
  float* cfB   = cfA + NN * 208;
  float* cfACC = cfB + NN * 208;
  float* cfBp  = cfACC + NN * 208;      // 480
  float* pBp   = cfBp + 480;            // 80
  float* cfWp  = pBp + 80;              // 1037568
  float* pWp   = cfWp + 1037568;        // 78336
  float* PA    = pWp + 78336;           // 384*384*32
  float* PB    = PA + (long)NN2 * 32;
  float* PACC  = PB + (long)NN2 * 32;

  // ---- prep: norm + dense S + padded weights/biases ----
  k_dinv<<<1, NN, 0, stream>>>(ei, ew, dinv);
  k_norm<<<(NE + 255) / 256, 256, 0, stream>>>(ei, ew, dinv, nrm);
  k_buildS<<<1, NN, 0, stream>>>(ei, nrm, S);

  long cfWoff[5], pWoff[4];
  int cfBoff[5], pBoff[4];
  {
    long o = 0;
    for (int l = 0; l < 5; ++l) { cfWoff[l] = o; o += (long)(KCF + 1) * cfCinP[l] * cfCoutP[l]; }
    o = 0;
    for (int l = 0; l < 4; ++l) { pWoff[l] = o; o += (long)(KP + 1) * pCinP[l] * pCoutP[l]; }
    int b = 0;
    for (int l = 0; l < 5; ++l) { cfBoff[l] = b; b += cfCoutP[l]; }
    b = 0;
    for (int l = 0; l < 4; ++l) { pBoff[l] = b; b += pCoutP[l]; }
  }
  for (int l = 0; l < 5; ++l) {
    int tot = (KCF + 1) * cfCinP[l] * cfCoutP[l];
    k_pad_w<<<(tot + 255) / 256, 256, 0, stream>>>((const float*)d_in[3 + l], cfWp + cfWoff[l],
                                                   KCF + 1, cfCin[l], cfCout[l], cfCinP[l], cfCoutP[l]);
    k_pad_b<<<1, 256, 0, stream>>>((const float*)d_in[8 + l], cfBp + cfBoff[l], cfCout[l], cfCoutP[l]);
  }
  for (int l = 0; l < 4; ++l) {
    int tot = (KP + 1) * pCinP[l] * pCoutP[l];
    k_pad_w<<<(tot + 255) / 256, 256, 0, stream>>>((const float*)d_in[13 + l], pWp + pWoff[l],
                                                   KP + 1, pCin[l], pCout[l], pCinP[l], pCoutP[l]);
    k_pad_b<<<1, 256, 0, stream>>>((const float*)d_in[17 + l], pBp + pBoff[l], pCout[l], pCoutP[l]);
  }

  // ---- CF network: 5 TAGConv layers over (384, C) ----
  k_cf_h0<<<(NN * 16 + 255) / 256, 256, 0, stream>>>(x, cfA);
  float* Hc = cfA;
  float* Hn = cfB;
  for (int l = 0; l < 5; ++l) {
    const int cin = cfCinP[l], cout = cfCoutP[l];
    const float* Wl = cfWp + cfWoff[l];
    const long kstride = (long)cin * cout;
    launch_gemm(Hc, Wl, cfACC, NN, cout, cin, false, stream);          // acc = h @ W0
    for (int k = 1; k <= KCF; ++k) {
      launch_gemm(S, Hc, Hn, NN, cin, NN, false, stream);              // h = S @ h
      launch_gemm(Hn, Wl + (long)k * kstride, cfACC, NN, cout, cin, true, stream);
      float* t = Hc; Hc = Hn; Hn = t;
    }
    if (l < 4) {
      k_bias_relu<<<(NN * cout + 255) / 256, 256, 0, stream>>>(cfACC, cfBp + cfBoff[l], Hn, NN, cout);
      float* t = Hc; Hc = Hn; Hn = t;                                  // relu output is next input
    } else {
      k_sigc<<<(NN + 255) / 256, 256, 0, stream>>>(cfACC, (const float*)d_in[12], cvec, cout);
    }
  }

  // ---- P network: 4 TAGConv layers over (384, 384, C) ----
  k_p_h0<<<(NN2 * 16 + 255) / 256, 256, 0, stream>>>(cvec, PA);
  float* Pc = PA;
  float* Pn = PB;
  for (int l = 0; l < 4; ++l) {
    const int cin = pCinP[l], cout = pCoutP[l];
    const float* Wl = pWp + pWoff[l];
    const long kstride = (long)cin * cout;
    // acc = h @ W0 : (N*N, cin) @ (cin, cout)
    launch_gemm(Pc, Wl, PACC, NN2, cout, cin, false, stream);
    for (int k = 1; k <= KP; ++k) {
      launch_gemm(S, Pc, Pn, NN, NN * cin, NN, false, stream);         // h = S @ h (wide GEMM)
      launch_gemm(Pn, Wl + (long)k * kstride, PACC, NN2, cout, cin, true, stream);
      float* t = Pc; Pc = Pn; Pn = t;
    }
    if (l < 3) {
      k_bias_relu<<<(NN2 * cout + 255) / 256, 256, 0, stream>>>(PACC, pBp + pBoff[l], Pn, NN2, cout);
      float* t = Pc; Pc = Pn; Pn = t;
    } else {
      k_final<<<(NN2 + 255) / 256, 256, 0, stream>>>(PACC, (const float*)d_in[20], cvec, out, cout);
    }
  }
}